// PCritical_46815143526805
// MI455X (gfx1250) — compile-verified
//
#include <hip/hip_runtime.h>
#include <hip/hip_bf16.h>

#define NN   4096
#define BB   32
#define TILE 256
#define NT   (NN / TILE)      // 16 tiles per dim, 16 partial slabs

#define ALPHA 0.025f
#define BETA  0.00025f
#define VTH   1.0f
#define TNOW  10.0f
#define INV_TAU_V 0.98019867330675525f   // exp(-1/50)
#define INV_TAU_I 0.36787944117144233f   // exp(-1)

typedef float v2f __attribute__((ext_vector_type(2)));
typedef float v8f __attribute__((ext_vector_type(8)));

static __device__ __forceinline__ v8f vzero8() {
    v8f z = {0.f, 0.f, 0.f, 0.f, 0.f, 0.f, 0.f, 0.f};
    return z;
}

// ---------------------------------------------------------------------------
// Kernel 1: per-neuron auxiliaries.
//   smask[n]  : bit b set iff mem_pot[b,n] > VTH  (spike S as a wave32 bitmask)
//   fg[n]     : 1.0 if any batch has S_paired spike (sp_max>0), else 0
//   eg[n]     : ALPHA * exp(max_st[n]/50)    rg[n] : exp(-max_st[n]/50)
// so that ALPHA*exp(|a_i-a_j|/50) = max(eg_i,eg_j) * max(rg_i,rg_j)
// ---------------------------------------------------------------------------
__global__ void k_aux(const float* __restrict__ mem_pot,
                      const float* __restrict__ mem_pot_p,
                      const float* __restrict__ st,
                      unsigned*    __restrict__ smask,
                      float* __restrict__ eg,
                      float* __restrict__ rg,
                      float* __restrict__ fg)
{
    const int n = blockIdx.x * blockDim.x + threadIdx.x;
    if (n >= NN) return;
    unsigned mask = 0u;
    int pany = 0;
    for (int b = 0; b < BB; ++b) {
        const float mp  = mem_pot  [(size_t)b * NN + n];
        const float mpp = mem_pot_p[(size_t)b * NN + n];
        if (mp  > VTH)          mask |= (1u << b);
        if (mpp > VTH + ALPHA)  pany  = 1;
    }
    const int cnt = __popc(mask);
    const float sn = st[n];
    // max over batch of where(S>0, T_NOW, st[n])
    const float a = (cnt == BB) ? TNOW : (cnt > 0 ? fmaxf(TNOW, sn) : sn);
    const float e = __expf(a * 0.02f);
    smask[n] = mask;
    eg[n] = ALPHA * e;
    rg[n] = 1.0f / e;
    fg[n] = pany ? 1.0f : 0.0f;
}

// ---------------------------------------------------------------------------
// Kernel 2: fused W_new computation + both GEMMs via f32 WMMA.
// Grid: (NT, NT) tiles of 256x256; 256 threads = 8 waves per WG.
// Wave w owns j-strips {2w, 2w+1} (16 cols each), sweeps all 16 i-strips.
//   R[b,j] partial (summed over this tile's 256 i)  -> pR[tI][b][j]
//   C[b,i] partial (summed over this tile's 256 j)  -> pC[tJ][b][i]
// ---------------------------------------------------------------------------
__global__ __launch_bounds__(256) void k_main(
    const float*    __restrict__ Wrec,
    const unsigned* __restrict__ smask,
    const float* __restrict__ eg,
    const float* __restrict__ rg,
    const float* __restrict__ fg,
    float* __restrict__ Wnew,
    float* __restrict__ pR,
    float* __restrict__ pC)
{
    __shared__ float ls_er[TILE], ls_rr[TILE], ls_fr[TILE];   // row aux (i-range)
    __shared__ float ls_ec[TILE], ls_rc[TILE];                // col aux (j-range)
    __shared__ float ls_w[8][16 * 17];                        // per-wave tile, pad 17
    __shared__ float ls_cred[8][BB][16];                      // cross-wave C reduce

    const int tJ = blockIdx.x, tI = blockIdx.y;
    const int tid  = threadIdx.x;
    const int wave = tid >> 5, lane = tid & 31;
    const int i0 = tI * TILE, j0 = tJ * TILE;

    ls_er[tid] = eg[i0 + tid];
    ls_rr[tid] = rg[i0 + tid];
    ls_fr[tid] = fg[i0 + tid];
    ls_ec[tid] = eg[j0 + tid];
    ls_rc[tid] = rg[j0 + tid];
    __syncthreads();

    const int colL  = lane & 15;   // WMMA "N" lane index (also batch index m)
    const int rhalf = lane >> 4;   // lane half selects K sub-pair
    const int kk    = rhalf * 2;

    v8f rAcc[2][2];                // [j-strip][batch-half], 32b x 16j each
    for (int s = 0; s < 2; ++s)
        for (int h = 0; h < 2; ++h) rAcc[s][h] = vzero8();

    for (int t = 0; t < 16; ++t) {          // i-strips, lockstep across waves
        const int ib = t * 16;
        v8f cAcc[2];                        // 32b x 16i, summed over wave's 32 j
        cAcc[0] = vzero8(); cAcc[1] = vzero8();

        for (int s = 0; s < 2; ++s) {
            const int jl = (wave * 2 + s) * 16;
            const int gj = j0 + jl;
            const float ecol = ls_ec[jl + colL];
            const float rcol = ls_rc[jl + colL];

            // ---- compute 16x16 W_new micro-tile (coalesced), stage in LDS ----
            for (int r = 0; r < 8; ++r) {
                const int row = 2 * r + rhalf;
                const size_t gidx = (size_t)(i0 + ib + row) * NN + gj + colL;
                const float w    = Wrec[gidx];
                const float erow = ls_er[ib + row];
                const float rrow = ls_rr[ib + row];
                const float frow = ls_fr[ib + row];
                // ALPHA*exp(|ai-aj|/50) = max(ALPHA*ei,ALPHA*ej)*max(ri,rj)
                const float pen = fmaxf(erow, ecol) * fmaxf(rrow, rcol) * frow;
                float u = (w + BETA) - pen;
                u = fminf(fmaxf(u, 0.0f), 1.0f);
                const float wn = (w > 0.0f) ? u : w;   // sign_mask select
                Wnew[gidx] = wn;
                ls_w[wave][row * 17 + colL] = wn;
            }

            // ---- R += S(32xK=i) * w(i x 16j) : 4 K-chunks of 4 ----
            for (int c = 0; c < 4; ++c) {
                v2f bmat;
                bmat.x = ls_w[wave][(c * 4 + kk + 0) * 17 + colL];
                bmat.y = ls_w[wave][(c * 4 + kk + 1) * 17 + colL];
                const unsigned m0 = smask[i0 + ib + c * 4 + kk + 0];
                const unsigned m1 = smask[i0 + ib + c * 4 + kk + 1];
                for (int h = 0; h < 2; ++h) {
                    const int bit = colL + h * 16;     // batch = colL + 16h
                    v2f amat;
                    amat.x = (float)((m0 >> bit) & 1u);
                    amat.y = (float)((m1 >> bit) & 1u);
                    rAcc[s][h] = __builtin_amdgcn_wmma_f32_16x16x4_f32(
                        false, amat, false, bmat, (short)0, rAcc[s][h],
                        false, false);
                }
            }

            // ---- C += S(32xK=j) * w^T(j x 16i) : transposed LDS reads ----
            for (int c = 0; c < 4; ++c) {
                v2f bmat;
                bmat.x = ls_w[wave][colL * 17 + c * 4 + kk + 0];
                bmat.y = ls_w[wave][colL * 17 + c * 4 + kk + 1];
                const unsigned m0 = smask[gj + c * 4 + kk + 0];
                const unsigned m1 = smask[gj + c * 4 + kk + 1];
                for (int h = 0; h < 2; ++h) {
                    const int bit = colL + h * 16;
                    v2f amat;
                    amat.x = (float)((m0 >> bit) & 1u);
                    amat.y = (float)((m1 >> bit) & 1u);
                    cAcc[h] = __builtin_amdgcn_wmma_f32_16x16x4_f32(
                        false, amat, false, bmat, (short)0, cAcc[h],
                        false, false);
                }
            }
        }

        // ---- cross-wave reduce C for this i-strip (fixed order, no atomics) --
        for (int h = 0; h < 2; ++h)
            for (int v = 0; v < 8; ++v) {
                const int b = h * 16 + rhalf * 8 + v;   // C/D VGPR layout
                ls_cred[wave][b][colL] = cAcc[h][v];
            }
        __syncthreads();
        for (int q = 0; q < 2; ++q) {
            const int oid = tid * 2 + q;                // 512 outputs / 256 thr
            const int b = oid >> 4, il = oid & 15;
            float sum = 0.0f;
            for (int w2 = 0; w2 < 8; ++w2) sum += ls_cred[w2][b][il];
            pC[(size_t)tJ * (BB * NN) + (size_t)b * NN + (i0 + ib + il)] = sum;
        }
        __syncthreads();
    }

    // ---- write R partials (full sum over this tile's 256 rows) ----
    for (int s = 0; s < 2; ++s)
        for (int h = 0; h < 2; ++h)
            for (int v = 0; v < 8; ++v) {
                const int b = h * 16 + rhalf * 8 + v;
                const int j = j0 + (wave * 2 + s) * 16 + colL;
                pR[(size_t)tI * (BB * NN) + (size_t)b * NN + j] = rAcc[s][h][v];
            }
}

// ---------------------------------------------------------------------------
// Kernel 3: reduce the NT partial slabs in fixed order; integrate/leak/reset.
// ---------------------------------------------------------------------------
__global__ void k_final(
    const float* __restrict__ inp,
    const float* __restrict__ mem_pot,
    const float* __restrict__ mem_cur,
    const float* __restrict__ mem_pot_p,
    const float* __restrict__ mem_cur_p,
    const int*   __restrict__ refrac,
    const float* __restrict__ pR,
    const float* __restrict__ pC,
    float* __restrict__ oS,   float* __restrict__ oPot,  float* __restrict__ oCur,
    float* __restrict__ oPotP,float* __restrict__ oCurP, float* __restrict__ oRef)
{
    const size_t idx = (size_t)blockIdx.x * blockDim.x + threadIdx.x;
    if (idx >= (size_t)BB * NN) return;

    float R = 0.0f, C = 0.0f;
    for (int p = 0; p < NT; ++p) {
        R += pR[(size_t)p * (BB * NN) + idx];
        C += pC[(size_t)p * (BB * NN) + idx];
    }
    const float mp  = mem_pot[idx];
    const float mpp = mem_pot_p[idx];
    const float S  = (mp  > VTH)         ? 1.0f : 0.0f;   // clip(ceil(mp-1),0,1)
    const float Sp = (mpp > VTH + ALPHA) ? 1.0f : 0.0f;

    int rf = refrac[idx];
    if (rf > 0) rf -= 1;
    const bool act = (rf == 0);

    float cur  = mem_cur[idx];
    float curp = mem_cur_p[idx];
    if (act) { cur = inp[idx] + R + cur; curp = C + curp; }
    float pot  = act ? (cur  + mp ) : mp;
    float potp = act ? (curp + mpp) : mpp;

    pot  *= INV_TAU_V;  cur  *= INV_TAU_I;
    potp *= INV_TAU_V;  curp *= INV_TAU_I;

    if (S  > 0.0f) pot  = 0.0f;
    if (Sp > 0.0f) potp = 0.0f;
    const float rnew = (S > 0.0f) ? 2.0f : (float)rf;

    oS[idx]    = S;    oPot[idx]  = pot;  oCur[idx]  = cur;
    oPotP[idx] = potp; oCurP[idx] = curp; oRef[idx]  = rnew;
}

// ---------------------------------------------------------------------------
extern "C" void kernel_launch(void* const* d_in, const int* in_sizes, int n_in,
                              void* d_out, int out_size, void* d_ws, size_t ws_size,
                              hipStream_t stream)
{
    (void)in_sizes; (void)n_in; (void)out_size; (void)ws_size;

    const float* inp   = (const float*)d_in[0];
    const float* Wrec  = (const float*)d_in[1];
    const float* mpot  = (const float*)d_in[2];
    const float* mcur  = (const float*)d_in[3];
    const float* mpotp = (const float*)d_in[4];
    const float* mcurp = (const float*)d_in[5];
    const float* st    = (const float*)d_in[6];
    const int*   refr  = (const int*)d_in[7];

    const size_t BN = (size_t)BB * NN;
    const size_t NW = (size_t)NN * NN;
    float* out = (float*)d_out;
    // return order: S, mem_pot, W_new, mem_cur, mem_pot_p, mem_cur_p, refrac
    float* oS    = out;
    float* oPot  = out + BN;
    float* oW    = out + 2 * BN;
    float* oCur  = out + 2 * BN + NW;
    float* oPotP = out + 3 * BN + NW;
    float* oCurP = out + 4 * BN + NW;
    float* oRef  = out + 5 * BN + NW;

    char* ws = (char*)d_ws;
    unsigned* smask = (unsigned*)(ws);
    float* eg = (float*)(ws + 16384);
    float* rg = (float*)(ws + 32768);
    float* fg = (float*)(ws + 49152);
    float* pR = (float*)(ws + 65536);           // NT * B * N floats (8 MB)
    float* pC = pR + (size_t)NT * BN;           // NT * B * N floats (8 MB)

    k_aux<<<NN / 256, 256, 0, stream>>>(mpot, mpotp, st, smask, eg, rg, fg);

    dim3 grid(NT, NT);
    k_main<<<grid, 256, 0, stream>>>(Wrec, smask, eg, rg, fg, oW, pR, pC);

    k_final<<<(unsigned)(BN / 256), 256, 0, stream>>>(
        inp, mpot, mcur, mpotp, mcurp, refr, pR, pC,
        oS, oPot, oCur, oPotP, oCurP, oRef);
}